// PackedINTAsymmetricWeightsDequantizer_15625091022917
// MI455X (gfx1250) — compile-verified
//
#include <hip/hip_runtime.h>
#include <stdint.h>

// Problem geometry (fixed by the reference).
#define OUT_FEATURES     8192
#define PACKED_COLS      4096                      // IN_FEATURES / 2, one byte per int32
#define THREADS          256                       // 8 wave32s per block
#define INTS_PER_THREAD  4                         // one b128 load per lane per chunk
#define CHUNK_INTS       (THREADS * INTS_PER_THREAD)   // 1024 int32 = 4 KB
#define NCHUNK           (PACKED_COLS / CHUNK_INTS)    // 4 chunks per row

typedef __attribute__((ext_vector_type(4))) int   int4v;
typedef __attribute__((ext_vector_type(4))) float float4v;

// Pointer types matching the async-copy builtin's parameters (revealed by the
// compiler diagnostics): arg0 = int4v in global AS(1), arg1 = int4v in LDS AS(3).
typedef __attribute__((address_space(1))) int4v global_int4v;
typedef __attribute__((address_space(3))) int4v lds_int4v;

// ---- gfx1250 async-copy path detection -------------------------------------
#if defined(__gfx1250__) && __has_builtin(__builtin_amdgcn_global_load_async_to_lds_b128)
#define USE_ASYNC 1
#else
#define USE_ASYNC 0
#endif

#if defined(__gfx1250__)
#if __has_builtin(__builtin_amdgcn_s_wait_asynccnt)
#define WAIT_ASYNC(n) __builtin_amdgcn_s_wait_asynccnt(n)
#else
#define WAIT_ASYNC(n) asm volatile("s_wait_asynccnt %0" :: "i"(n) : "memory")
#endif
#else
#define WAIT_ASYNC(n)
#endif

// Dequantize 4 packed int32 (4 bytes -> 8 nibbles) into 8 floats and store
// with non-temporal hint (output is 256 MB streamed once; > 192 MB L2).
__device__ __forceinline__ void dequant8_store(int4v w, float s, float b,
                                               float* __restrict__ dst) {
    float4v o0, o1;
    o0.x = fmaf((float)( w.x        & 0xF), s, b);
    o0.y = fmaf((float)((w.x >> 4)  & 0xF), s, b);
    o0.z = fmaf((float)( w.y        & 0xF), s, b);
    o0.w = fmaf((float)((w.y >> 4)  & 0xF), s, b);
    o1.x = fmaf((float)( w.z        & 0xF), s, b);
    o1.y = fmaf((float)((w.z >> 4)  & 0xF), s, b);
    o1.z = fmaf((float)( w.w        & 0xF), s, b);
    o1.w = fmaf((float)((w.w >> 4)  & 0xF), s, b);
    __builtin_nontemporal_store(o0, (float4v*)dst);
    __builtin_nontemporal_store(o1, (float4v*)(dst + 4));
}

__global__ __launch_bounds__(THREADS)
void PackedINT4_dequant_kernel(const int*   __restrict__ weight,
                               const float* __restrict__ scale,
                               const float* __restrict__ zero_point,
                               float*       __restrict__ out) {
    const int row = blockIdx.x;          // one output row per block
    const int t   = threadIdx.x;

    // Per-row affine params are block-uniform -> scalarized by the compiler.
    const float s = scale[row];
    const float b = -zero_point[row] * s;   // (w - zp)*s == fma(w, s, -zp*s)

    const int*   wrow = weight + (size_t)row * PACKED_COLS;
    float*       orow = out    + (size_t)row * (size_t)(2 * PACKED_COLS);

#if USE_ASYNC
    // Double-buffered async Global->LDS staging. Each lane consumes only the
    // 16 B it requested itself, so per-wave ASYNCcnt ordering is sufficient
    // (no workgroup barriers). 8 KB LDS per block (WGP has 320 KB).
    __shared__ __align__(16) int lds[2][CHUNK_INTS];

    // Prime the pipeline with chunk 0.
    __builtin_amdgcn_global_load_async_to_lds_b128(
        (global_int4v*)(wrow + t * INTS_PER_THREAD),
        (lds_int4v*)&lds[0][t * INTS_PER_THREAD],
        /*offset=*/0, /*cpol=*/0);

    for (int c = 0; c < NCHUNK; ++c) {
        if (c + 1 < NCHUNK) {
            // Kick off the next chunk before consuming the current one.
            __builtin_amdgcn_global_load_async_to_lds_b128(
                (global_int4v*)(wrow + (c + 1) * CHUNK_INTS + t * INTS_PER_THREAD),
                (lds_int4v*)&lds[(c + 1) & 1][t * INTS_PER_THREAD],
                /*offset=*/0, /*cpol=*/0);
            WAIT_ASYNC(1);   // chunk c complete (async loads retire in order)
        } else {
            WAIT_ASYNC(0);   // drain for the final chunk
        }

        int4v w = *(const int4v*)&lds[c & 1][t * INTS_PER_THREAD];
        dequant8_store(w, s, b,
                       orow + (size_t)2 * ((size_t)c * CHUNK_INTS
                                           + (size_t)t * INTS_PER_THREAD));
    }
#else
    // Fallback: direct NT b128 loads + prefetch of the next chunk
    // (global_prefetch_b8 on gfx1250).
    for (int c = 0; c < NCHUNK; ++c) {
        const int* gp = wrow + c * CHUNK_INTS + t * INTS_PER_THREAD;
        if (c + 1 < NCHUNK)
            __builtin_prefetch(gp + CHUNK_INTS, 0, 0);
        int4v w = __builtin_nontemporal_load((const int4v*)gp);
        dequant8_store(w, s, b,
                       orow + (size_t)2 * ((size_t)c * CHUNK_INTS
                                           + (size_t)t * INTS_PER_THREAD));
    }
#endif
}

extern "C" void kernel_launch(void* const* d_in, const int* in_sizes, int n_in,
                              void* d_out, int out_size, void* d_ws, size_t ws_size,
                              hipStream_t stream) {
    const int*   weight     = (const int*)d_in[0];     // [8192, 4096] int32 (byte values)
    const float* scale      = (const float*)d_in[1];   // [8192, 1]
    const float* zero_point = (const float*)d_in[2];   // [8192, 1]
    float*       out        = (float*)d_out;           // [8192, 8192] float32

    dim3 grid(OUT_FEATURES);
    dim3 block(THREADS);
    PackedINT4_dequant_kernel<<<grid, block, 0, stream>>>(weight, scale, zero_point, out);
}